// SpectralTokenTimeAttention_27960237097492
// MI455X (gfx1250) — compile-verified
//
#include <hip/hip_runtime.h>
#include <hip/hip_bf16.h>
#include <math.h>

// ---------------- problem constants ----------------
constexpr int B_  = 16;
constexpr int N_  = 4096;
constexpr int D_  = 64;
constexpr int O_  = 12;
constexpr int S_  = 32;
constexpr int M_  = 12;   // K_TOD + K_DOW tokens
constexpr int Mp  = 16;   // m padded to 16 for WMMA
constexpr int A_  = 32;   // attn dim
constexpr float ALPHA  = 0.1f;
constexpr float SCALE  = 0.17677669529663687f;  // 1/sqrt(32), TEMP=1
constexpr float TWO_PI = 6.28318530717958647692f;

constexpr int OMP = O_ * Mp;   // 192 padded (o,m) columns
constexpr int NT  = 64;        // N rows per block
constexpr int HP  = D_ + 4;    // 68: LDS row stride for H / Wkey (bank-conflict pad)
constexpr int LP  = OMP + 4;   // 196: LDS row stride for logits/attn
constexpr int VP  = Mp + 4;    // 20: LDS row stride for transposed val [o*64+d][m]
constexpr int WV_ELTS = O_ * D_ * VP;  // 15360 floats; >= OMP*HP = 13056 (aliased)

typedef __attribute__((ext_vector_type(2))) float v2f;
typedef __attribute__((ext_vector_type(8))) float v8f;

// =====================================================================
// Kernel A: per (b,o) fold the token-side math into
//   Wkey[b,o,mP,d] = sum_a Wq[a,d] * key[b,o,m,a]     (mP zero-padded 12..15)
//   bias[b,o,mP]   = sum_a qs[b,o,a] * key[b,o,m,a]
//   valT[b,o,d,mP] = gate[b,o,m] * val_emb[m,d]       (TRANSPOSED, zero-padded)
// and emit the gate output directly.
// =====================================================================
__global__ __launch_bounds__(64)
void stta_setup_kernel(const float* __restrict__ ts_out,    // [B,O,2]
                       const float* __restrict__ step_emb,  // [B,O,S]
                       const float* __restrict__ key_emb,   // [M,A]
                       const float* __restrict__ val_emb,   // [M,D]
                       const float* __restrict__ Wk,        // [A,2]
                       const float* __restrict__ Wg,        // [1,2]
                       const float* __restrict__ bg,        // [1]
                       const float* __restrict__ Wq,        // [A, D+S]
                       const float* __restrict__ bq,        // [A]
                       float* __restrict__ wkey_g,          // [B,O,Mp,D]
                       float* __restrict__ valt_g,          // [B,O,D,Mp]
                       float* __restrict__ bias_g,          // [B,O,Mp]
                       float* __restrict__ gate_out)        // [B,O,M]
{
    const int bo = blockIdx.x;     // b*O + o
    const int t  = threadIdx.x;    // 0..63

    __shared__ float s_tok[M_][2];
    __shared__ float s_key[M_][A_];
    __shared__ float s_qs[A_];
    __shared__ float s_gate[M_];

    const float tod = ts_out[bo * 2 + 0];
    const float dow = ts_out[bo * 2 + 1];

    if (t < M_) {
        const int   m     = t;
        const float phase = (m < 8) ? tod : dow;
        const int   k     = (m < 8) ? (m + 1) : (m - 8 + 1);
        const float ang   = TWO_PI * phase * (float)k;
        const float s = sinf(ang);
        const float c = cosf(ang);
        s_tok[m][0] = s;
        s_tok[m][1] = c;
        const float g = tanhf(s * Wg[0] + c * Wg[1] + bg[0]);   // GATE_BOUND = 1
        s_gate[m] = g;
        gate_out[bo * M_ + m] = g;
    }
    if (t < A_) {
        float acc = bq[t];
        const float* se = step_emb + (size_t)bo * S_;
        const float* wr = Wq + (size_t)t * (D_ + S_) + D_;
        #pragma unroll 8
        for (int s2 = 0; s2 < S_; ++s2) acc += se[s2] * wr[s2];
        s_qs[t] = acc;
    }
    __syncthreads();

    for (int idx = t; idx < M_ * A_; idx += 64) {
        const int m = idx / A_, a = idx % A_;
        s_key[m][a] = s_tok[m][0] * Wk[a * 2 + 0] +
                      s_tok[m][1] * Wk[a * 2 + 1] + key_emb[m * A_ + a];
    }
    __syncthreads();

    if (t < Mp) {
        float acc = 0.0f;
        if (t < M_) {
            #pragma unroll 8
            for (int a = 0; a < A_; ++a) acc += s_qs[a] * s_key[t][a];
        }
        bias_g[(size_t)bo * Mp + t] = acc;
    }

    // Wkey, [Mp][D] with zero pad rows
    for (int idx = t; idx < Mp * D_; idx += 64) {
        const int m = idx / D_, d = idx % D_;
        float wv = 0.0f;
        if (m < M_) {
            #pragma unroll 8
            for (int a = 0; a < A_; ++a) wv += Wq[a * (D_ + S_) + d] * s_key[m][a];
        }
        wkey_g[(size_t)bo * Mp * D_ + idx] = wv;
    }
    // transposed val, [D][Mp] with zero pad cols (m=12..15)
    for (int idx = t; idx < D_ * Mp; idx += 64) {
        const int d = idx >> 4, m = idx & 15;
        const float vv = (m < M_) ? s_gate[m] * val_emb[m * D_ + d] : 0.0f;
        valt_g[(size_t)bo * D_ * Mp + idx] = vv;
    }
}

// =====================================================================
// Kernel B: per (b, 64-row N tile):
//   phase 1: logits = H_tile(16x64) x Wkey^T(64x192) via v_wmma_f32_16x16x4_f32
//   phase 2: softmax over m=12 per (row,o); write attn; zero-pad m=12..15;
//            restage valT[b] into the (now dead) Wkey LDS region
//   phase 3: delta = attn(16x16) x val(16x64) via WMMA (all-LDS operands),
//            fused H_time = H + 0.1*delta epilogue
// =====================================================================
__global__ __launch_bounds__(128)
void stta_main_kernel(const float* __restrict__ H,        // [B,N,D]
                      const float* __restrict__ wkey_g,   // [B,O,Mp,D]
                      const float* __restrict__ valt_g,   // [B,O,D,Mp]
                      const float* __restrict__ bias_g,   // [B,O,Mp]
                      float* __restrict__ htime_out,      // [B,O,N,D]
                      float* __restrict__ attn_out)       // [B,O,N,M]
{
    __shared__ float sH[NT * HP];     // H tile, row stride 68
    __shared__ float sWV[WV_ELTS];    // phase1: Wkey [om][d] stride 68 (13056 used)
                                      // phase3: valT [o*64+d][m] stride 20 (aliased)
    __shared__ float sL[NT * LP];     // logits -> attn, row stride 196

    const int b    = blockIdx.y;
    const int n0   = blockIdx.x * NT;
    const int t    = threadIdx.x;
    const int lane = t & 31;
    const int wave = t >> 5;
    const int rowl = lane & 15;
    const int half = lane >> 4;       // 0: lanes 0-15, 1: lanes 16-31

    // ---- stage H tile (64x64 f32) and Wkey[b] (192x64 f32) in LDS ----
    {
        const float4* src = (const float4*)(H + ((size_t)b * N_ + n0) * D_);
        for (int i = t; i < NT * (D_ / 4); i += 128) {
            const int r = i >> 4, c4 = i & 15;
            *(float4*)&sH[r * HP + c4 * 4] = src[r * 16 + c4];
        }
    }
    {
        const float4* src = (const float4*)(wkey_g + (size_t)b * OMP * D_);
        for (int i = t; i < OMP * (D_ / 4); i += 128) {
            const int r = i >> 4, c4 = i & 15;
            *(float4*)&sWV[r * HP + c4 * 4] = src[r * 16 + c4];
        }
    }
    __syncthreads();

    // ---- phase 1: logits GEMM, 16 rows/wave, 12 col tiles, K=64 ----
    {
        const int rbase = wave * 16;
        v2f afr[16];
        #pragma unroll
        for (int kk = 0; kk < 16; ++kk)
            afr[kk] = *(const v2f*)&sH[(rbase + rowl) * HP + 4 * kk + 2 * half];

        for (int o = 0; o < O_; ++o) {           // col tile == o (16 cols each)
            v8f acc = {};
            #pragma unroll
            for (int kk = 0; kk < 16; ++kk) {
                const v2f bfr =
                    *(const v2f*)&sWV[(o * 16 + rowl) * HP + 4 * kk + 2 * half];
                acc = __builtin_amdgcn_wmma_f32_16x16x4_f32(
                        false, afr[kk], false, bfr, (short)0, acc, false, false);
            }
            const int   col = o * 16 + rowl;
            const float bv  = bias_g[(size_t)b * OMP + col];
            #pragma unroll
            for (int v = 0; v < 8; ++v) {
                const int r = rbase + v + 8 * half;
                sL[r * LP + col] = (acc[v] + bv) * SCALE;
            }
        }
    }
    __syncthreads();   // sL valid everywhere; sWV (Wkey) now dead

    // ---- phase 2a: restage valT[b] into sWV, rows [o*64+d], stride 20 ----
    {
        const float4* src = (const float4*)(valt_g + (size_t)b * O_ * D_ * Mp);
        for (int i = t; i < O_ * D_ * (Mp / 4); i += 128) {   // 3072 float4s
            const int row = i >> 2, c4 = i & 3;               // row = o*64+d
            *(float4*)&sWV[row * VP + c4 * 4] = src[i];       // 20%4==0 -> aligned
        }
    }

    // ---- phase 2b: softmax over m (12) for each (row, o) pair ----
    for (int p = t; p < NT * O_; p += 128) {
        const int r = p & (NT - 1);
        const int o = p >> 6;
        float* lrow = &sL[r * LP + o * Mp];
        float mx = lrow[0];
        #pragma unroll
        for (int m = 1; m < M_; ++m) mx = fmaxf(mx, lrow[m]);
        float e[M_];
        float sum = 0.0f;
        #pragma unroll
        for (int m = 0; m < M_; ++m) { e[m] = __expf(lrow[m] - mx); sum += e[m]; }
        const float inv = 1.0f / sum;
        const size_t abase = (((size_t)b * O_ + o) * N_ + (n0 + r)) * M_;
        #pragma unroll
        for (int m = 0; m < M_; ++m) {
            const float a = e[m] * inv;
            lrow[m] = a;
            attn_out[abase + m] = a;
        }
        #pragma unroll
        for (int m = M_; m < Mp; ++m) lrow[m] = 0.0f;   // pad for WMMA K
    }
    __syncthreads();

    // ---- phase 3: delta = attn x val via WMMA (LDS-only operands) ----
    {
        const int rbase = wave * 16;
        for (int o = 0; o < O_; ++o) {
            v2f afr[4];
            #pragma unroll
            for (int kk = 0; kk < 4; ++kk)
                afr[kk] = *(const v2f*)
                    &sL[(rbase + rowl) * LP + o * Mp + 4 * kk + 2 * half];

            const size_t obase = (((size_t)b * O_ + o) * N_ + n0) * D_;

            #pragma unroll
            for (int ct = 0; ct < 4; ++ct) {     // 4 tiles of 16 d-columns
                const int d = ct * 16 + rowl;
                v8f acc = {};
                #pragma unroll
                for (int kk = 0; kk < 4; ++kk) {
                    const v2f bfr = *(const v2f*)
                        &sWV[(o * D_ + d) * VP + 4 * kk + 2 * half];
                    acc = __builtin_amdgcn_wmma_f32_16x16x4_f32(
                            false, afr[kk], false, bfr, (short)0, acc,
                            false, false);
                }
                #pragma unroll
                for (int v = 0; v < 8; ++v) {
                    const int r = rbase + v + 8 * half;
                    htime_out[obase + (size_t)r * D_ + d] =
                        sH[r * HP + d] + ALPHA * acc[v];
                }
            }
        }
    }
}

// =====================================================================
// host launcher
// =====================================================================
extern "C" void kernel_launch(void* const* d_in, const int* in_sizes, int n_in,
                              void* d_out, int out_size, void* d_ws, size_t ws_size,
                              hipStream_t stream) {
    const float* H        = (const float*)d_in[0];
    const float* ts_out   = (const float*)d_in[1];
    const float* step_emb = (const float*)d_in[2];
    const float* key_emb  = (const float*)d_in[3];
    const float* val_emb  = (const float*)d_in[4];
    const float* Wk       = (const float*)d_in[5];
    const float* Wg       = (const float*)d_in[6];
    const float* bg       = (const float*)d_in[7];
    const float* Wq       = (const float*)d_in[8];
    const float* bq       = (const float*)d_in[9];

    float* out = (float*)d_out;
    float* htime_out = out;                                          // [B,O,N,D]
    float* attn_out  = out + (size_t)B_ * O_ * N_ * D_;              // [B,O,N,M]
    float* gate_out  = attn_out + (size_t)B_ * O_ * N_ * M_;         // [B,O,M]

    // workspace: Wkey | valT | bias  (~1.6 MB)
    float* wkey_g = (float*)d_ws;                                    // [B,O,Mp,D]
    float* valt_g = wkey_g + (size_t)B_ * O_ * Mp * D_;              // [B,O,D,Mp]
    float* bias_g = valt_g + (size_t)B_ * O_ * D_ * Mp;              // [B,O,Mp]

    stta_setup_kernel<<<dim3(B_ * O_), dim3(64), 0, stream>>>(
        ts_out, step_emb, key_emb, val_emb, Wk, Wg, bg, Wq, bq,
        wkey_g, valt_g, bias_g, gate_out);

    stta_main_kernel<<<dim3(N_ / NT, B_), dim3(128), 0, stream>>>(
        H, wkey_g, valt_g, bias_g, htime_out, attn_out);
}